// CTCFsdPrefixSearch_67310727463188
// MI455X (gfx1250) — compile-verified
//
#include <hip/hip_runtime.h>
#include <math.h>

#define NEGV (-1e35f)

typedef __attribute__((ext_vector_type(2))) float v2f;
typedef __attribute__((ext_vector_type(8))) float v8f;

#if defined(__has_builtin)
#if __has_builtin(__builtin_amdgcn_wmma_f32_16x16x4_f32)
#define HAVE_WMMA_F32X4 1
#endif
#endif
#ifndef HAVE_WMMA_F32X4
#define HAVE_WMMA_F32X4 0
#endif

// ---------------------------------------------------------------------------
// Kernel 1: per-row logsumexp of logits [T, V].
// One 256-thread block (8 waves) per 16-row tile. Wave w owns the 4-column
// groups starting at 4*w with stride 32. Within a wave, lane L covers row
// (L&15) and columns {2*(L>>4), 2*(L>>4)+1} of each group -- exactly the
// 32-bit A-matrix 16x4 WMMA layout (lanes 0-15: K=0,1; lanes 16-31: K=2,3).
// Pass 1: per-wave row max -> LDS -> block row max.
// Pass 2 (L2-resident re-read): exp via v_exp_f32 (__expf), summed by
// V_WMMA_F32_16X16X4_F32 against an all-ones B so the 16 row-sums ride the
// XDL f32 accumulator; per-wave partials combined through LDS.
// ---------------------------------------------------------------------------
__global__ __launch_bounds__(256)
void lse_rows_kernel(const float* __restrict__ logits, float* __restrict__ lse,
                     int T, int V) {
  const int tid = threadIdx.x;
  const int w = tid >> 5;     // wave id 0..7
  const int lane = tid & 31;
  const int m = lane & 15;    // row within 16-row tile
  const int g = lane >> 4;    // column half: cols {2g, 2g+1} within a group
  const int t0 = blockIdx.x << 4;
  const int row = t0 + m;
  const int rowc = (row < T) ? row : (T - 1);
  const float* rp = logits + (size_t)rowc * (size_t)V;
  const int Vfloor = V & ~3;

  __shared__ float smax[8][16];
  __shared__ float ssum[8][16];

  // ---- pass 1: row max over this wave's column stripe ----
  float mx = -INFINITY;
  for (int cb = 4 * w; cb < Vfloor; cb += 32) {
    int c0 = cb + 2 * g;
    float x0 = rp[c0];
    float x1 = rp[c0 + 1];
    mx = fmaxf(mx, fmaxf(x0, x1));
  }
  if (w == 0) {  // tail columns (V % 4) handled by wave 0
    int c0 = Vfloor + 2 * g;
    if (c0 < V)     mx = fmaxf(mx, rp[c0]);
    if (c0 + 1 < V) mx = fmaxf(mx, rp[c0 + 1]);
  }
  mx = fmaxf(mx, __shfl_xor(mx, 16, 32));  // combine column halves
  if (lane < 16) smax[w][m] = mx;
  __syncthreads();

  float rowmax = smax[0][m];
#pragma unroll
  for (int ww = 1; ww < 8; ++ww) rowmax = fmaxf(rowmax, smax[ww][m]);

  // ---- pass 2: exp-sum (stripe) ----
  float tsum = 0.f;  // tail contribution (wave 0 only)
#if HAVE_WMMA_F32X4
  v8f acc = {0.f, 0.f, 0.f, 0.f, 0.f, 0.f, 0.f, 0.f};
  v2f ones; ones[0] = 1.0f; ones[1] = 1.0f;
#else
  float sum = 0.f;
#endif
  for (int cb = 4 * w; cb < Vfloor; cb += 32) {
    int c0 = cb + 2 * g;
    v2f a;
    a[0] = __expf(rp[c0] - rowmax);       // x - rowmax <= 0: safe for fast exp
    a[1] = __expf(rp[c0 + 1] - rowmax);
#if HAVE_WMMA_F32X4
    // D[m][n] = sum_k A[m][k] * 1 + C[m][n]  -> row-sums broadcast over n
    acc = __builtin_amdgcn_wmma_f32_16x16x4_f32(
        /*neg_a=*/false, a, /*neg_b=*/false, ones,
        /*c_mod=*/(short)0, acc, /*reuse_a=*/false, /*reuse_b=*/false);
#else
    sum += a[0] + a[1];
#endif
  }
  if (w == 0) {
    int c0 = Vfloor + 2 * g;
    if (c0 < V)     tsum += __expf(rp[c0] - rowmax);
    if (c0 + 1 < V) tsum += __expf(rp[c0 + 1] - rowmax);
  }
  tsum += __shfl_xor(tsum, 16, 32);  // per-row tail total (both halves hold it)

#if HAVE_WMMA_F32X4
  // C layout: VGPR r -> lane 0 holds row r (r=0..7), lane 16 holds row 8+r.
#pragma unroll
  for (int r = 0; r < 8; ++r) {
    float ts_lo = __shfl(tsum, r, 32);
    float ts_hi = __shfl(tsum, 8 + r, 32);
    if (lane == 0)  ssum[w][r]     = acc[r] + ts_lo;
    if (lane == 16) ssum[w][8 + r] = acc[r] + ts_hi;
  }
#else
  sum += tsum;
  sum += __shfl_xor(sum, 16, 32);
  if (lane < 16) ssum[w][m] = sum;
#endif
  __syncthreads();

  if (tid < 16) {
    float tot = 0.f;
#pragma unroll
    for (int ww = 0; ww < 8; ++ww) tot += ssum[ww][tid];
    int r = t0 + tid;
    if (r < T) lse[r] = rowmax + __logf(tot);
  }
}

// ---------------------------------------------------------------------------
// Kernel 2: gather E[t][u] = logits[t, sym[u]] - lse[t], sym[u] = odd? tgt[u/2] : 0
// ---------------------------------------------------------------------------
__global__ void gather_e_kernel(const float* __restrict__ logits,
                                const int* __restrict__ tgt,
                                const float* __restrict__ lse,
                                float* __restrict__ E,
                                int T, int V, int Uphi) {
  int idx = blockIdx.x * blockDim.x + threadIdx.x;
  if (idx >= T * Uphi) return;
  int t = idx / Uphi;
  int u = idx - t * Uphi;
  int sym = (u & 1) ? tgt[u >> 1] : 0;
  E[idx] = logits[(size_t)t * (size_t)V + (size_t)sym] - lse[t];
}

// ---------------------------------------------------------------------------
// Kernel 3: sequential CTC recursion. Single wave32, 2 states per lane:
// lane L owns states L ("lo") and 32+L ("hi"). Neighbor states (u-1, u-2)
// come via cross-lane shuffles (no LDS barriers). Next E row is prefetched
// while the current step computes.
// ---------------------------------------------------------------------------
__device__ __forceinline__ void ctc_update(int u, float a, float s, float c,
                                           float a1, float s1, float c1,
                                           float a2, float s2, float c2,
                                           float e, float tf,
                                           float& na, float& ns, float& nc) {
  if (u & 1) {  // label state
    float c0v = a2 + e, c1v = a1 + e, c2v = a + e;
    bool p0 = (c0v >= c1v) && (c0v >= c2v);
    bool p1 = (!p0) && (c1v >= c2v);
    na = p0 ? c0v : (p1 ? c1v : c2v);
    ns = p0 ? s2 : (p1 ? s1 : s);
    nc = (p0 ? c2 : (p1 ? c1 : c)) + 1.0f;
  } else {      // blank state
    bool pe = (a >= a1);
    na = (pe ? a : a1) + e;
    ns = pe ? s : s1;
    nc = (pe ? c : c1) + 1.0f;
  }
  if (u < 2) { na = 0.0f; ns = tf; nc = 1.0f; }
}

__global__ __launch_bounds__(32)
void ctc_scan_kernel(const float* __restrict__ E, float* __restrict__ out,
                     int T, int Uphi) {
  const int L = threadIdx.x;
  const int uLo = L;
  const int uHi = 32 + L;
  const bool loValid = (uLo < Uphi);
  const bool hiValid = (uHi < Uphi);

  float aLo = (uLo < 2) ? 0.f : NEGV;
  float sLo = (uLo < 2) ? 0.f : -1.f;
  float cLo = (uLo < 2) ? 1.f : 0.f;
  float aHi = NEGV, sHi = -1.f, cHi = 0.f;  // uHi >= 32 > 1 always

  float* oa = out + 1;
  float* os = out + 1 + (size_t)T;
  float* oc = out + 1 + 2 * (size_t)T;

  // pick state q out of the distributed (lo,hi) register pair
  auto pick = [&](int q, float lo, float hi) -> float {
    float vlo = __shfl(lo, q & 31, 32);
    float vhi = __shfl(hi, (q - 32) & 31, 32);
    return (q < 32) ? vlo : vhi;
  };

  const int qL = Uphi - 1;
  const int qP = (Uphi >= 2) ? (Uphi - 2) : 0;

  // row 0 output (pick over initial state)
  {
    float aL = pick(qL, aLo, aHi), aP = pick(qP, aLo, aHi);
    float sL = pick(qL, sLo, sHi), sP = pick(qP, sLo, sHi);
    float cL = pick(qL, cLo, cHi), cP = pick(qP, cLo, cHi);
    bool tl = (aL >= aP);
    if (L == 0) {
      float va = tl ? aL : aP, vs = tl ? sL : sP, vc = tl ? cL : cP;
      oa[0] = va; os[0] = vs; oc[0] = vc;
      out[0] = va;  // covers T==1; overwritten at t==T-1 otherwise
    }
  }

  float eLo = 0.f, eHi = 0.f;
  if (T > 1) {
    eLo = loValid ? E[uLo] : 0.f;
    eHi = hiValid ? E[uHi] : 0.f;
  }

  for (int t = 1; t < T; ++t) {
    // prefetch E row t (used at iteration t+1, which consumes E[(t+1)-1])
    float eLoN = 0.f, eHiN = 0.f;
    if (t + 1 < T) {
      eLoN = loValid ? E[(size_t)t * Uphi + uLo] : 0.f;
      eHiN = hiValid ? E[(size_t)t * Uphi + uHi] : 0.f;
    }

    // lo-state neighbors (u-1, u-2)
    float a1Lo = __shfl_up(aLo, 1, 32), s1Lo = __shfl_up(sLo, 1, 32), c1Lo = __shfl_up(cLo, 1, 32);
    float a2Lo = __shfl_up(aLo, 2, 32), s2Lo = __shfl_up(sLo, 2, 32), c2Lo = __shfl_up(cLo, 2, 32);
    if (L < 1) { a1Lo = NEGV; s1Lo = 0.f; c1Lo = 0.f; }
    if (L < 2) { a2Lo = NEGV; s2Lo = 0.f; c2Lo = 0.f; }

    // hi-state neighbors cross the lane-31/state-32 boundary
    float aT30 = __shfl(aLo, 30, 32), sT30 = __shfl(sLo, 30, 32), cT30 = __shfl(cLo, 30, 32);
    float aT31 = __shfl(aLo, 31, 32), sT31 = __shfl(sLo, 31, 32), cT31 = __shfl(cLo, 31, 32);
    float a1Hi = __shfl_up(aHi, 1, 32), s1Hi = __shfl_up(sHi, 1, 32), c1Hi = __shfl_up(cHi, 1, 32);
    float a2Hi = __shfl_up(aHi, 2, 32), s2Hi = __shfl_up(sHi, 2, 32), c2Hi = __shfl_up(cHi, 2, 32);
    if (L == 0) { a1Hi = aT31; s1Hi = sT31; c1Hi = cT31;
                  a2Hi = aT30; s2Hi = sT30; c2Hi = cT30; }
    else if (L == 1) { a2Hi = aT31; s2Hi = sT31; c2Hi = cT31; }

    float tf = (float)t;
    float naLo, nsLo, ncLo, naHi, nsHi, ncHi;
    ctc_update(uLo, aLo, sLo, cLo, a1Lo, s1Lo, c1Lo, a2Lo, s2Lo, c2Lo, eLo, tf,
               naLo, nsLo, ncLo);
    ctc_update(uHi, aHi, sHi, cHi, a1Hi, s1Hi, c1Hi, a2Hi, s2Hi, c2Hi, eHi, tf,
               naHi, nsHi, ncHi);
    if (!hiValid) { naHi = NEGV; nsHi = 0.f; ncHi = 0.f; }
    aLo = naLo; sLo = nsLo; cLo = ncLo;
    aHi = naHi; sHi = nsHi; cHi = ncHi;

    // pick_out over states (Uphi-1, Uphi-2), ties -> last
    float aL = pick(qL, aLo, aHi), aP = pick(qP, aLo, aHi);
    float sL2 = pick(qL, sLo, sHi), sP = pick(qP, sLo, sHi);
    float cL2 = pick(qL, cLo, cHi), cP = pick(qP, cLo, cHi);
    bool tl = (aL >= aP);
    if (L == 0) {
      float va = tl ? aL : aP, vs = tl ? sL2 : sP, vc = tl ? cL2 : cP;
      oa[t] = va; os[t] = vs; oc[t] = vc;
      if (t == T - 1) out[0] = va;  // forward_logprob
    }

    eLo = eLoN; eHi = eHiN;
  }
}

// ---------------------------------------------------------------------------
// Host launcher
// ---------------------------------------------------------------------------
extern "C" void kernel_launch(void* const* d_in, const int* in_sizes, int n_in,
                              void* d_out, int out_size, void* d_ws, size_t ws_size,
                              hipStream_t stream) {
  const float* logits = (const float*)d_in[0];   // [1, T, V] f32
  const int* targets = (const int*)d_in[1];      // [1, U] i32
  (void)n_in; (void)ws_size;

  const int U = in_sizes[1];
  const int T = (out_size - 1) / 3;              // out = 1 + 3*T floats
  const int V = (T > 0) ? (int)((size_t)in_sizes[0] / (size_t)T) : 0;
  const int Uphi = 2 * U + 1;

  float* lse = (float*)d_ws;                     // [T]
  float* E = lse + T;                            // [T, Uphi]

  if (T <= 0) return;

  lse_rows_kernel<<<(T + 15) / 16, 256, 0, stream>>>(logits, lse, T, V);

  long tot = (long)T * (long)Uphi;
  gather_e_kernel<<<(int)((tot + 255) / 256), 256, 0, stream>>>(
      logits, targets, lse, E, T, V, Uphi);

  ctc_scan_kernel<<<1, 32, 0, stream>>>(E, (float*)d_out, T, Uphi);
}